// FasterSeamlessM4TFeatureExtractor_42219528520235
// MI455X (gfx1250) — compile-verified
//
#include <hip/hip_runtime.h>
#include <math.h>

// ---------------------------------------------------------------------------
// SeamlessM4T feature extractor on gfx1250 via FP32 WMMA DFT + mel matmuls.
// B operands are pre-packed into WMMA lane layout so the hot loops issue one
// global_load_b128 (DFT) / global_load_b64 (mel) per WMMA step.
// ---------------------------------------------------------------------------

typedef float f32x2 __attribute__((ext_vector_type(2)));
typedef float f32x4 __attribute__((ext_vector_type(4)));
typedef float f32x8 __attribute__((ext_vector_type(8)));

#define FRAME_LEN   400
#define HOP         160
#define NBINS       257
#define NBIN_TILES  17           // 272 bins padded = 17*16
#define NMEL        80           // 5 * 16
#define NMEL_TILES  5
#define KSTEPS_DFT  100          // 400 / 4
#define KSTEPS_MEL  68           // 272 / 4
#define PREEMPH     0.97f
#define MEL_FLOOR   1.192092955078125e-07f
#define PRE_STRIDE  404          // LDS stride for prepped frames (conflict pad)
#define SPEC_STRIDE 276          // LDS stride for power spectrum (conflict pad)

// WMMA f32 16x16x4 lane mapping (ISA 7.12.2):
//   A (16x4): lane&15 = M; VGPR0 = K=k (lanes 0-15) / k+2 (lanes 16-31),
//             VGPR1 = k+1 / k+3.  B (4x16) mirrors with N = lane&15.

// --------------------------------------------------------------------------
// Kernel 1a: DFT basis packed in B-operand lane layout.
//   bPk[((t*100 + i)*32 + lane)*4 + {0,1,2,3}] =
//     { cos(n0,bin), cos(n0+1,bin), sin(n0,bin), sin(n0+1,bin) }
//   n0 = i*4 + koff(lane), bin = t*16 + (lane&15); zero for bin >= 257.
// Integer phase reduction mod 512 keeps fp32 sincos exact-argument.
// --------------------------------------------------------------------------
__global__ void build_basis_packed(float* __restrict__ bPk) {
    int idx = blockIdx.x * blockDim.x + threadIdx.x;           // slot id
    if (idx >= NBIN_TILES * KSTEPS_DFT * 32) return;
    const int lane = idx & 31;
    const int rest = idx >> 5;
    const int i = rest % KSTEPS_DFT;
    const int t = rest / KSTEPS_DFT;
    const int bin  = t * 16 + (lane & 15);
    const int koff = (lane < 16) ? 0 : 2;
    const int n0 = i * 4 + koff;

    f32x4 v = {0.f, 0.f, 0.f, 0.f};
    if (bin < NBINS) {
        float s0, c0, s1, c1;
        int p0 = (n0 * bin) & 511;
        int p1 = ((n0 + 1) * bin) & 511;
        sincosf((6.283185307179586f / 512.0f) * (float)p0, &s0, &c0);
        sincosf((6.283185307179586f / 512.0f) * (float)p1, &s1, &c1);
        v.x = c0; v.y = c1; v.z = s0; v.w = s1;   // sin sign irrelevant (|X|^2)
    }
    *(f32x4*)(bPk + (size_t)idx * 4) = v;
}

// --------------------------------------------------------------------------
// Kernel 1b: mel filter bank packed in B-operand lane layout (K padded 272).
//   mPk[((tile*68 + i)*32 + lane)*2 + {0,1}] = { W[kr][mn], W[kr+1][mn] }
// --------------------------------------------------------------------------
__global__ void build_mel_packed(const float* __restrict__ melW,
                                 float* __restrict__ mPk) {
    int idx = blockIdx.x * blockDim.x + threadIdx.x;
    if (idx >= NMEL_TILES * KSTEPS_MEL * 32) return;
    const int lane = idx & 31;
    const int rest = idx >> 5;
    const int i = rest % KSTEPS_MEL;
    const int tile = rest / KSTEPS_MEL;
    const int mn   = tile * 16 + (lane & 15);
    const int koff = (lane < 16) ? 0 : 2;
    const int kr = i * 4 + koff;

    f32x2 v = {0.f, 0.f};
    if (kr     < NBINS) v.x = melW[kr * NMEL + mn];
    if (kr + 1 < NBINS) v.y = melW[(kr + 1) * NMEL + mn];
    *(f32x2*)(mPk + (size_t)idx * 2) = v;
}

// --------------------------------------------------------------------------
// Kernel 2: per 16-frame strip: prep -> DFT (WMMA f32) -> |X|^2 -> mel (WMMA)
// Block = 256 threads = 8 waves; all intermediates stay in LDS.
// --------------------------------------------------------------------------
__global__ __launch_bounds__(256)
void dft_mel(const float* __restrict__ wav,
             const float* __restrict__ window,
             const float* __restrict__ bPk,       // packed DFT basis
             const float* __restrict__ mPk,       // packed mel weights
             float* __restrict__ melOut,          // [F][80]
             int F) {
    __shared__ float preBuf[16 * PRE_STRIDE];
    __shared__ float specBuf[16 * SPEC_STRIDE];

    const int tid  = threadIdx.x;
    const int lane = tid & 31;
    const int wave = tid >> 5;
    const int frameBase = blockIdx.x * 16;

    // ---- Phase 1: frame prep (scale, DC removal, preemphasis, window) ----
    for (int fr = wave; fr < 16; fr += 8) {
        const int g = frameBase + fr;
        const bool valid = (g < F);
        const long base = (long)g * HOP;

        float sum = 0.f;
        if (valid) {
            for (int k = lane; k < FRAME_LEN; k += 32)
                sum += wav[base + k];
        }
        sum += __shfl_xor(sum, 16, 32);
        sum += __shfl_xor(sum,  8, 32);
        sum += __shfl_xor(sum,  4, 32);
        sum += __shfl_xor(sum,  2, 32);
        sum += __shfl_xor(sum,  1, 32);
        const float mean = sum * (32768.0f / (float)FRAME_LEN);

        for (int k = lane; k < FRAME_LEN; k += 32) {
            float v = 0.f;
            if (valid) {
                float x = wav[base + k] * 32768.0f;
                if (k == 0) {
                    v = (x - mean) * (1.0f - PREEMPH);
                } else {
                    float xm = wav[base + k - 1] * 32768.0f;
                    v = x - PREEMPH * xm - (1.0f - PREEMPH) * mean;
                }
                v *= window[k];
            }
            preBuf[fr * PRE_STRIDE + k] = v;
        }
    }
    __syncthreads();

    const int mrow = lane & 15;
    const int koff = (lane < 16) ? 0 : 2;

    // ---- Phase 2: DFT + power spectrum, 17 bin-tiles over 8 waves ----
    for (int t = wave; t < NBIN_TILES; t += 8) {
        const int bin = t * 16 + mrow;
        const f32x4* pB =
            (const f32x4*)bPk + ((size_t)t * KSTEPS_DFT) * 32 + lane;

        f32x8 accC = {0.f, 0.f, 0.f, 0.f, 0.f, 0.f, 0.f, 0.f};
        f32x8 accS = {0.f, 0.f, 0.f, 0.f, 0.f, 0.f, 0.f, 0.f};

        for (int i = 0; i < KSTEPS_DFT; ++i) {
            f32x4 b = pB[(size_t)i * 32];                 // one b128 per step
            f32x2 a = *(const f32x2*)&preBuf[mrow * PRE_STRIDE + i * 4 + koff];
            f32x2 vc = {b.x, b.y};
            f32x2 vs = {b.z, b.w};
            accC = __builtin_amdgcn_wmma_f32_16x16x4_f32(
                false, a, false, vc, (short)0, accC, false, false);
            accS = __builtin_amdgcn_wmma_f32_16x16x4_f32(
                false, a, false, vs, (short)0, accS, false, false);
        }

        // C/D layout: VGPR r, lanes 0-15 -> M=r, lanes 16-31 -> M=8+r
        #pragma unroll
        for (int r = 0; r < 8; ++r) {
            const int frame = (lane < 16) ? r : (8 + r);
            const float re = accC[r];
            const float im = accS[r];
            specBuf[frame * SPEC_STRIDE + bin] = re * re + im * im;
        }
    }
    __syncthreads();

    // ---- Phase 3: mel projection [16 x 272] @ [272 x 80] + log, 5 tiles ----
    if (wave < NMEL_TILES) {
        const int mn = wave * 16 + mrow;
        const f32x2* pM =
            (const f32x2*)mPk + ((size_t)wave * KSTEPS_MEL) * 32 + lane;

        f32x8 acc = {0.f, 0.f, 0.f, 0.f, 0.f, 0.f, 0.f, 0.f};

        for (int i = 0; i < KSTEPS_MEL; ++i) {
            f32x2 b = pM[(size_t)i * 32];                 // one b64 per step
            f32x2 a = *(const f32x2*)&specBuf[mrow * SPEC_STRIDE + i * 4 + koff];
            acc = __builtin_amdgcn_wmma_f32_16x16x4_f32(
                false, a, false, b, (short)0, acc, false, false);
        }

        #pragma unroll
        for (int r = 0; r < 8; ++r) {
            const int row = frameBase + ((lane < 16) ? r : (8 + r));
            if (row < F)
                melOut[(size_t)row * NMEL + mn] =
                    logf(fmaxf(acc[r], MEL_FLOOR));
        }
    }
}

// --------------------------------------------------------------------------
// Kernel 3: deterministic per-column mean / unbiased-var (f64 accumulation).
// --------------------------------------------------------------------------
__global__ __launch_bounds__(256)
void col_stats(const float* __restrict__ mel, float* __restrict__ stats, int F) {
    __shared__ double sS[256];
    __shared__ double sQ[256];
    const int c = blockIdx.x;
    double s = 0.0, q = 0.0;
    for (int f = threadIdx.x; f < F; f += 256) {
        double v = (double)mel[(size_t)f * NMEL + c];
        s += v;
        q += v * v;
    }
    sS[threadIdx.x] = s;
    sQ[threadIdx.x] = q;
    __syncthreads();
    for (int off = 128; off > 0; off >>= 1) {
        if (threadIdx.x < off) {
            sS[threadIdx.x] += sS[threadIdx.x + off];
            sQ[threadIdx.x] += sQ[threadIdx.x + off];
        }
        __syncthreads();
    }
    if (threadIdx.x == 0) {
        double mean = sS[0] / (double)F;
        double var  = (sQ[0] - sS[0] * sS[0] / (double)F) / (double)(F - 1);
        stats[c * 2 + 0] = (float)mean;
        stats[c * 2 + 1] = (float)(1.0 / sqrt(var + 1e-7));
    }
}

// --------------------------------------------------------------------------
// Kernel 4: normalize -> output
// --------------------------------------------------------------------------
__global__ void norm_out(const float* __restrict__ mel,
                         const float* __restrict__ stats,
                         float* __restrict__ out, int N) {
    int i = blockIdx.x * blockDim.x + threadIdx.x;
    if (i >= N) return;
    int c = i % NMEL;
    out[i] = (mel[i] - stats[c * 2 + 0]) * stats[c * 2 + 1];
}

// --------------------------------------------------------------------------
extern "C" void kernel_launch(void* const* d_in, const int* in_sizes, int n_in,
                              void* d_out, int out_size, void* d_ws, size_t ws_size,
                              hipStream_t stream) {
    const float* wav    = (const float*)d_in[0];   // [T]
    const float* melW   = (const float*)d_in[1];   // [257*80]
    const float* window = (const float*)d_in[2];   // [400]

    const long T = (long)in_sizes[0];
    const int  F = 1 + (int)((T - FRAME_LEN) / HOP);

    // Workspace layout (floats)
    float* ws     = (float*)d_ws;
    float* bPk    = ws;                                        // 17*100*32*4
    float* mPk    = bPk + (size_t)NBIN_TILES * KSTEPS_DFT * 32 * 4;  // 5*68*32*2
    float* melBuf = mPk + (size_t)NMEL_TILES * KSTEPS_MEL * 32 * 2;  // F*80
    float* stats  = melBuf + (size_t)F * NMEL;                 // 160

    const int nb = NBIN_TILES * KSTEPS_DFT * 32;
    build_basis_packed<<<(nb + 255) / 256, 256, 0, stream>>>(bPk);

    const int nm = NMEL_TILES * KSTEPS_MEL * 32;
    build_mel_packed<<<(nm + 255) / 256, 256, 0, stream>>>(melW, mPk);

    const int ftiles = (F + 15) / 16;
    dft_mel<<<ftiles, 256, 0, stream>>>(wav, window, bPk, mPk, melBuf, F);

    col_stats<<<NMEL, 256, 0, stream>>>(melBuf, stats, F);

    const int N = F * NMEL;
    norm_out<<<(N + 255) / 256, 256, 0, stream>>>(melBuf, stats, (float*)d_out, N);
}